// WeightedAveragePoints_45784351375411
// MI455X (gfx1250) — compile-verified
//
#include <hip/hip_runtime.h>

// WeightedAveragePoints on gfx1250 (MI455X):
//   out[b,k,j] = sum_n w[b,k,n]*p[b,j,n] / max(sum_n w[b,k,n], 1e-12)
// B=64, K=8, J=3, N=76800. HBM-bound (~216MB reads, ~9.3us floor @ 23.3TB/s).
// Pipeline: async global->LDS (ASYNCcnt) double-buffered tiles, reduction via
// V_WMMA_F32_16X16X4_F32 with a ones-column so D[k,3] accumulates the
// denominator in the same matrix op.

#define B_   64
#define K_   8
#define J_   3
#define SMAX 30
#define TILE 512
#define RS   (TILE + 4)   // padded LDS row stride (floats) -> conflict-free A reads

typedef __attribute__((ext_vector_type(2))) float v2f;
typedef __attribute__((ext_vector_type(4))) float v4f;
typedef __attribute__((ext_vector_type(8))) float v8f;
typedef __attribute__((ext_vector_type(4))) int   i4;

#if defined(__has_builtin)
#  if __has_builtin(__builtin_amdgcn_global_load_async_to_lds_b128) && \
      __has_builtin(__builtin_amdgcn_s_wait_asynccnt)
#    define HAVE_ASYNC 1
#  endif
#endif
#ifndef HAVE_ASYNC
#  define HAVE_ASYNC 0
#endif

typedef __attribute__((address_space(1))) i4 gi4;  // global int4
typedef __attribute__((address_space(3))) i4 li4;  // LDS int4

__device__ __forceinline__ void cp_b128(const float* __restrict__ g, float* l) {
#if HAVE_ASYNC
  // global_load_async_to_lds_b128: memory -> LDS, no VGPR round trip, ASYNCcnt.
  __builtin_amdgcn_global_load_async_to_lds_b128(
      (gi4*)g, (li4*)l, /*imm offset*/ 0, /*cpol*/ 0);
#else
  *(v4f*)l = *(const v4f*)g;  // sync fallback through VGPRs
#endif
}

__device__ __forceinline__ void wait_async() {
#if HAVE_ASYNC
  __builtin_amdgcn_s_wait_asynccnt(0);
#endif
}

__global__ __launch_bounds__(256) void wap_partial(
    const float* __restrict__ points, const float* __restrict__ weights,
    float* __restrict__ ws, int N, int S)
{
  __shared__ float wT[2][K_ * RS];   // weights tile, double buffered
  __shared__ float pT[2][4 * RS];    // points rows 0..2 + ones row 3
  __shared__ float sacc[8 * 32];

  const int tid  = threadIdx.x;
  const int lane = tid & 31;
  const int wave = tid >> 5;
  const int sIdx = blockIdx.x;
  const int b    = blockIdx.y;

  const float* wBase = weights + (size_t)b * K_ * N;
  const float* pBase = points  + (size_t)b * J_ * N;

  // Ones rows (denominator column), written once per buffer; staging only
  // overwrites rows 0..2.
  for (int q = tid; q < TILE / 4; q += 256) {
    v4f one = {1.f, 1.f, 1.f, 1.f};
    *(v4f*)&pT[0][3 * RS + q * 4] = one;
    *(v4f*)&pT[1][3 * RS + q * 4] = one;
  }

  // ---- Per-thread staging descriptors (computed once) ----
  // weights: 8 rows x 128 float4 = 1024 chunks -> 4 per thread
  const float* gw0[4];
  int lw[4], wq4[4];
  #pragma unroll
  for (int r = 0; r < 4; ++r) {
    const int idx = r * 256 + tid;
    const int k = idx >> 7, q = idx & 127;
    gw0[r] = wBase + (size_t)k * N + q * 4;
    lw[r]  = k * RS + q * 4;
    wq4[r] = q * 4;
  }
  // points: 3 rows x 128 float4 = 384 chunks -> r=0 all threads, r=1 tid<128
  const float* gp0[2];
  int lp[2], pq4[2];
  #pragma unroll
  for (int r = 0; r < 2; ++r) {
    const int idx = r * 256 + tid;
    const int j = idx >> 7, q = idx & 127;
    gp0[r] = pBase + (size_t)j * N + q * 4;
    lp[r]  = j * RS + q * 4;
    pq4[r] = q * 4;
  }
  const bool p1ok = (256 + tid) < J_ * (TILE / 4);
  const bool nAligned = (N & 3) == 0;

  auto stage = [&](int base, int buf) {
    if (nAligned && base + TILE <= N) {         // fast path: full tile, async
      #pragma unroll
      for (int r = 0; r < 4; ++r) cp_b128(gw0[r] + base, &wT[buf][lw[r]]);
      cp_b128(gp0[0] + base, &pT[buf][lp[0]]);
      if (p1ok) cp_b128(gp0[1] + base, &pT[buf][lp[1]]);
    } else {                                    // guarded tail, zero padded
      #pragma unroll
      for (int r = 0; r < 4; ++r) {
        const int n = base + wq4[r];
        const float* g = gw0[r] + base;
        float t0 = (n + 0 < N) ? g[0] : 0.f, t1 = (n + 1 < N) ? g[1] : 0.f;
        float t2 = (n + 2 < N) ? g[2] : 0.f, t3 = (n + 3 < N) ? g[3] : 0.f;
        *(v4f*)&wT[buf][lw[r]] = (v4f){t0, t1, t2, t3};
      }
      #pragma unroll
      for (int r = 0; r < 2; ++r) {
        if (r == 1 && !p1ok) break;
        const int n = base + pq4[r];
        const float* g = gp0[r] + base;
        float t0 = (n + 0 < N) ? g[0] : 0.f, t1 = (n + 1 < N) ? g[1] : 0.f;
        float t2 = (n + 2 < N) ? g[2] : 0.f, t3 = (n + 3 < N) ? g[3] : 0.f;
        *(v4f*)&pT[buf][lp[r]] = (v4f){t0, t1, t2, t3};
      }
    }
  };

  // WMMA fragment addressing (wave32), V_WMMA_F32_16X16X4_F32:
  //  A 16x4: lanes 0-15 row M=lane K={0,1}; lanes 16-31 K={2,3}.
  //  B 4x16: lanes 0-15 col N=lane K={0,1}; lanes 16-31 K={2,3}.
  // Rows M>=8 alias weight rows 0..7, cols N>=4 alias point rows 0..3 so no
  // lane diverges (EXEC must be all ones); garbage D rows/cols never read.
  const int arow = lane & 7;
  const int brow = lane & 3;
  const int koff = (lane >> 4) * 2;
  const int nl0  = wave * (TILE / 8);

  v8f acc = {};

  const int base0 = sIdx * TILE;
  const int step  = S * TILE;

  if (base0 < N) stage(base0, 0);   // prologue: fill buffer 0
  wait_async();
  __syncthreads();

  int buf = 0;
  for (int base = base0; base < N; base += step) {
    const int nextBase = base + step;
    if (nextBase < N) stage(nextBase, buf ^ 1);   // overlap with compute

    // Warm L2 two tiles ahead (global_prefetch_b8, speculative).
    const int pfBase = base + 2 * step;
    if (pfBase + TILE <= N) {
      __builtin_prefetch(gw0[0] + pfBase, 0, 1);
      __builtin_prefetch(gp0[0] + pfBase, 0, 1);
    }

    const float* wr = &wT[buf][arow * RS];
    const float* pr = &pT[buf][brow * RS];
    #pragma unroll
    for (int i = 0; i < TILE / 8 / 4; ++i) {  // 16 WMMAs per wave per tile
      const int nl = nl0 + i * 4 + koff;
      v2f a  = *(const v2f*)&wr[nl];  // ds_load_b64, 8B aligned
      v2f bb = *(const v2f*)&pr[nl];
      acc = __builtin_amdgcn_wmma_f32_16x16x4_f32(
          /*neg_a=*/false, a, /*neg_b=*/false, bb,
          /*c_mod=*/(short)0, acc, /*reuse_a=*/false, /*reuse_b=*/false);
    }

    wait_async();      // my async writes to buf^1 complete
    __syncthreads();   // all waves' writes visible, all reads of buf done
    buf ^= 1;
  }

  // D layout: VGPR i, lanes 0-15 -> (M=i, N=lane). Extract rows 0..7 (k),
  // cols 0..3 (num_xyz, den); fixed-order cross-wave reduction (no atomics).
  if (lane < 4) {
    #pragma unroll
    for (int k = 0; k < 8; ++k) sacc[wave * 32 + k * 4 + lane] = acc[k];
  }
  __syncthreads();
  if (tid < 32) {
    float s = 0.f;
    #pragma unroll
    for (int w = 0; w < 8; ++w) s += sacc[w * 32 + tid];
    ws[((size_t)b * S + sIdx) * 32 + tid] = s;
  }
}

__global__ void wap_finalize(const float* __restrict__ ws, float* __restrict__ out, int S) {
  const int idx = blockIdx.x * blockDim.x + threadIdx.x;
  if (idx >= B_ * K_ * J_) return;
  const int b = idx / (K_ * J_);
  const int r = idx - b * (K_ * J_);
  const int k = r / J_;
  const int j = r - k * J_;
  const float* p = ws + (size_t)b * S * 32 + k * 4;
  float num = 0.f, den = 0.f;
  for (int s = 0; s < S; ++s) {  // fixed order -> bitwise deterministic
    num += p[(size_t)s * 32 + j];
    den += p[(size_t)s * 32 + 3];
  }
  out[idx] = num / fmaxf(den, 1e-12f);
}

extern "C" void kernel_launch(void* const* d_in, const int* in_sizes, int n_in,
                              void* d_out, int out_size, void* d_ws, size_t ws_size,
                              hipStream_t stream) {
  const float* points  = (const float*)d_in[0];   // [64,3,240,320] f32
  const float* weights = (const float*)d_in[1];   // [64,8,240,320] f32
  float* out = (float*)d_out;                     // [64,8,3] f32
  float* ws  = (float*)d_ws;

  const int N = in_sizes[0] / (B_ * J_);          // 76800

  // Partial buffer needs 64*S*32 floats; clamp S to workspace capacity.
  int S = SMAX;
  const size_t perS = (size_t)B_ * 32 * sizeof(float);
  if (ws_size < (size_t)S * perS) {
    S = (int)(ws_size / perS);
    if (S < 1) S = 1;
  }

  dim3 grid(S, B_);
  wap_partial<<<grid, dim3(256), 0, stream>>>(points, weights, ws, N, S);

  const int total = B_ * K_ * J_;
  wap_finalize<<<(total + 255) / 256, dim3(256), 0, stream>>>(ws, out, S);
}